// HungarianMatcher_81939386073356
// MI455X (gfx1250) — compile-verified
//
#include <hip/hip_runtime.h>

// Hungarian matcher (DETR-style) for gfx1250 / MI455X.
// One 256-thread workgroup (8 wave32) per batch. Cost matrix + all LSAP
// state live in LDS (~90KB of the 320KB/WGP). Boxes staged via the CDNA5
// Tensor Data Mover; union outer-sum term computed on the matrix pipe via
// v_wmma_f32_16x16x4_f32; Dijkstra argmin via wave32 shuffle butterflies.

#define N_PRED 300
#define M_TGT  64
#define NPAD   304
#define BLOCK  256
#define BIGF   1e30f

typedef float v2f  __attribute__((ext_vector_type(2)));
typedef float v8f  __attribute__((ext_vector_type(8)));
typedef unsigned int u32x4 __attribute__((ext_vector_type(4)));
typedef int   i32x4 __attribute__((ext_vector_type(4)));
typedef int   i32x8 __attribute__((ext_vector_type(8)));

#if __has_builtin(__builtin_amdgcn_tensor_load_to_lds)
#define HAVE_TDM 1
// 1-D global->LDS DMA of `ndw` dwords. D# layout per CDNA5 ISA:
// group0: count=1 | lds_addr | global_addr[56:0] | type=2
// group1: data_size=4B, tensor_dim0 = tile_dim0 = ndw, tensor_dim1 = 1.
// Higher-dimension groups (2/3/extra) unused for a 1-D copy -> zero.
__device__ __forceinline__ void tdm_load_1d(unsigned lds_off,
                                            const void* gptr,
                                            unsigned ndw) {
  unsigned long long ga = (unsigned long long)gptr;
  u32x4 g0;
  g0[0] = 1u;                                    // count = 1 (valid descriptor)
  g0[1] = lds_off;                               // lds_addr (bytes)
  g0[2] = (unsigned)ga;                          // global_addr[31:0]
  g0[3] = (unsigned)(ga >> 32) | (2u << 30);     // global_addr[56:32] | type=2
  i32x8 g1;
  g1[0] = (int)(2u << 16);                       // data_size = 2 -> 4 bytes
  g1[1] = (int)((ndw & 0xFFFFu) << 16);          // tensor_dim0 lo16 (bits 63:48)
  g1[2] = (int)((ndw >> 16) | (1u << 16));       // tensor_dim0 hi16 | tensor_dim1=1 lo16
  g1[3] = (int)((ndw & 0xFFFFu) << 16);          // tensor_dim1 hi16=0 | tile_dim0
  g1[4] = 0;                                     // tile_dim1=0, tile_dim2=0 (1-D)
  g1[5] = (int)ndw;                              // tensor_dim0_stride lo32
  g1[6] = 0;                                     // stride0 hi16 | stride1 lo16
  g1[7] = 0;                                     // stride1 hi32
  i32x4 gz4 = {0, 0, 0, 0};                      // groups 2/3 unused (dims <= 2)
  i32x8 gz8 = {0, 0, 0, 0, 0, 0, 0, 0};          // trailing group unused
  __builtin_amdgcn_tensor_load_to_lds(g0, g1, gz4, gz4, gz8, 0);
}
#endif

__global__ __launch_bounds__(BLOCK)
void hungarian_gfx1250(const float* __restrict__ pred,   // [B,300,4]
                       const float* __restrict__ tgt,    // [B,64,4]
                       const int*   __restrict__ mask,   // [B,64]
                       int* __restrict__ out,            // [2,B,64]
                       int B) {
  __shared__ float s_cost[M_TGT * NPAD];   // Ct[m][n], rows = targets
  __shared__ float s_pred[N_PRED * 4];
  __shared__ float s_tgt[M_TGT * 4];
  __shared__ float s_areap[NPAD];
  __shared__ float s_areat[M_TGT];
  __shared__ float s_u[M_TGT];
  __shared__ float s_v[NPAD];
  __shared__ float s_short[NPAD];
  __shared__ int   s_path[NPAD];
  __shared__ int   s_r4c[NPAD];            // row4col
  __shared__ int   s_sc[NPAD];             // SC flags
  __shared__ int   s_c4r[M_TGT];           // col4row
  __shared__ int   s_valid[M_TGT];
  __shared__ float s_redv[BLOCK / 32];
  __shared__ int   s_redi[BLOCK / 32];
  __shared__ int   s_i, s_sink;
  __shared__ float s_minv;

  const int tid  = threadIdx.x;
  const int b    = blockIdx.x;
  const int lane = tid & 31;
  const int wid  = tid >> 5;

  // ---- Stage boxes global -> LDS (Tensor Data Mover on gfx1250) ----
#if defined(HAVE_TDM)
  if (tid == 0) {
    tdm_load_1d((unsigned)(unsigned long long)&s_pred[0],
                pred + (size_t)b * N_PRED * 4, N_PRED * 4);
    tdm_load_1d((unsigned)(unsigned long long)&s_tgt[0],
                tgt + (size_t)b * M_TGT * 4, M_TGT * 4);
  }
  __builtin_amdgcn_s_wait_tensorcnt(0);    // no-op for waves with TENSORcnt==0
#else
  for (int k = tid; k < N_PRED * 4; k += BLOCK)
    s_pred[k] = pred[(size_t)b * N_PRED * 4 + k];
  for (int k = tid; k < M_TGT * 4; k += BLOCK)
    s_tgt[k] = tgt[(size_t)b * M_TGT * 4 + k];
#endif
  __syncthreads();

  // ---- Areas + validity ----
  for (int n = tid; n < NPAD; n += BLOCK) {
    float a = 0.0f;
    if (n < N_PRED) {
      const float* p = &s_pred[n * 4];
      a = (p[2] - p[0]) * (p[3] - p[1]);
    }
    s_areap[n] = a;
  }
  for (int m = tid; m < M_TGT; m += BLOCK) {
    const float* t = &s_tgt[m * 4];
    s_areat[m] = (t[2] - t[0]) * (t[3] - t[1]);
    s_valid[m] = mask[(size_t)b * M_TGT + m] > 0;
  }
  __syncthreads();

  // ---- Cost matrix, tile-parallel across waves ----
  // Union base area_t[m] + area_p[n] is a rank-2 outer sum == K=2 matmul:
  // [area_t,1] x [[1],[area_p]] done on the matrix pipe (wmma f32 16x16x4),
  // co-executing with the VALU L1/intersection math.
  {
    const int laneq = lane & 15;
    const int hi    = lane >> 4;               // 0: lanes 0-15, 1: lanes 16-31
    const int NT    = (NPAD / 16) * (M_TGT / 16);   // 19*4 = 76 tiles
    for (int t = wid; t < NT; t += BLOCK / 32) {
      const int m0 = (t & 3) << 4;
      const int n0 = (t >> 2) << 4;
      v2f av, bv;
      av[0] = hi ? 0.0f : s_areat[m0 + laneq];  // A[i,k]: k=0 -> area_t, k=1 -> 1
      av[1] = hi ? 0.0f : 1.0f;                 // lanes 16-31 carry k=2,3 -> 0
      bv[0] = hi ? 0.0f : 1.0f;                 // B[k,j]: k=0 -> 1, k=1 -> area_p
      bv[1] = hi ? 0.0f : s_areap[n0 + laneq];
      v8f acc = {};
      acc = __builtin_amdgcn_wmma_f32_16x16x4_f32(
          false, av, false, bv, (short)0, acc, false, false);

      const int n  = n0 + laneq;
      const int nc = (n < N_PRED) ? n : (N_PRED - 1);
      const float px1 = s_pred[nc * 4 + 0], py1 = s_pred[nc * 4 + 1];
      const float px2 = s_pred[nc * 4 + 2], py2 = s_pred[nc * 4 + 3];
#pragma unroll
      for (int r = 0; r < 8; ++r) {
        const int m = m0 + r + (hi << 3);       // C/D layout: VGPR r -> row r(+8)
        const float tx1 = s_tgt[m * 4 + 0], ty1 = s_tgt[m * 4 + 1];
        const float tx2 = s_tgt[m * 4 + 2], ty2 = s_tgt[m * 4 + 3];
        float l1 = fabsf(px1 - tx1) + fabsf(py1 - ty1) +
                   fabsf(px2 - tx2) + fabsf(py2 - ty2);
        float iw = fmaxf(fminf(px2, tx2) - fmaxf(px1, tx1), 0.0f);
        float ih = fmaxf(fminf(py2, ty2) - fmaxf(py1, ty1), 0.0f);
        float inter = iw * ih;
        float uni   = acc[r] - inter;           // area_t + area_p - inter
        float iou   = inter / (uni + 1e-7f);
        float c     = 5.0f * l1 - 2.0f * iou;
        c = s_valid[m] ? c : 0.0f;              // invalid target rows -> 0
        s_cost[m * NPAD + n] = (n < N_PRED) ? c : BIGF;
      }
    }
  }

  // ---- LSAP init ----
  for (int n = tid; n < NPAD; n += BLOCK) { s_v[n] = 0.0f; s_r4c[n] = -1; }
  for (int m = tid; m < M_TGT; m += BLOCK) { s_u[m] = 0.0f; s_c4r[m] = -1; }
  __syncthreads();

  // ---- Shortest augmenting path, one row (target) at a time ----
  for (int row = 0; row < M_TGT; ++row) {
    for (int n = tid; n < N_PRED; n += BLOCK) {
      s_short[n] = BIGF; s_path[n] = -1; s_sc[n] = 0;
    }
    if (tid == 0) { s_i = row; s_sink = -1; s_minv = 0.0f; }
    __syncthreads();

    for (;;) {
      const int   i  = s_i;
      const float mv = s_minv;
      const float ui = s_u[i];
      float bestv = __builtin_inff();
      int   bestn = 0x7fffffff;
      for (int n = tid; n < N_PRED; n += BLOCK) {
        if (!s_sc[n]) {
          float r = mv + s_cost[i * NPAD + n] - ui - s_v[n];
          if (r < s_short[n]) { s_short[n] = r; s_path[n] = i; }
          float val = s_short[n];
          if (val < bestv) { bestv = val; bestn = n; }   // ascending n: lowest idx on tie
        }
      }
      // wave32 argmin butterfly (lowest index wins ties, matching jnp.argmin)
#pragma unroll
      for (int off = 16; off > 0; off >>= 1) {
        float ov = __shfl_xor(bestv, off, 32);
        int   oi = __shfl_xor(bestn, off, 32);
        if (ov < bestv || (ov == bestv && oi < bestn)) { bestv = ov; bestn = oi; }
      }
      if (lane == 0) { s_redv[wid] = bestv; s_redi[wid] = bestn; }
      __syncthreads();
      if (tid == 0) {
        float bvv = s_redv[0]; int bj = s_redi[0];
        for (int w = 1; w < BLOCK / 32; ++w) {
          if (s_redv[w] < bvv || (s_redv[w] == bvv && s_redi[w] < bj)) {
            bvv = s_redv[w]; bj = s_redi[w];
          }
        }
        s_sc[bj] = 1;
        s_minv   = bvv;
        int rj = s_r4c[bj];
        if (rj < 0) s_sink = bj; else s_i = rj;
      }
      __syncthreads();
      if (s_sink >= 0) break;
    }

    // Dual updates: each SC column maps to a distinct assigned row -> parallel.
    const float mvf = s_minv;
    for (int n = tid; n < N_PRED; n += BLOCK) {
      if (s_sc[n]) {
        float d = mvf - s_short[n];
        s_v[n] -= d;
        int rn = s_r4c[n];
        if (rn >= 0) s_u[rn] += d;    // rn != row (row is unassigned)
      }
    }
    if (tid == 0) s_u[row] += mvf;
    __syncthreads();

    // Augment along alternating path (scalar pointer chase).
    if (tid == 0) {
      int j = s_sink;
      for (;;) {
        int pi = s_path[j];
        s_r4c[j] = pi;
        int jn = s_c4r[pi];
        s_c4r[pi] = j;
        if (pi == row) break;
        j = jn;
      }
    }
    __syncthreads();
  }

  // ---- Outputs: pred_idx [B,64] then tgt_idx [B,64] (int32) ----
  if (tid < M_TGT) {
    const int m  = tid;
    const int va = s_valid[m];
    out[(size_t)b * M_TGT + m]                      = va ? s_c4r[m] : -1;
    out[(size_t)B * M_TGT + (size_t)b * M_TGT + m]  = va ? m : -1;
  }
}

extern "C" void kernel_launch(void* const* d_in, const int* in_sizes, int n_in,
                              void* d_out, int out_size, void* d_ws, size_t ws_size,
                              hipStream_t stream) {
  (void)n_in; (void)out_size; (void)d_ws; (void)ws_size;
  const float* pred = (const float*)d_in[0];
  const float* tgt  = (const float*)d_in[1];
  const int*   mask = (const int*)d_in[2];
  int* out = (int*)d_out;
  const int B = in_sizes[0] / (N_PRED * 4);
  hipLaunchKernelGGL(hungarian_gfx1250, dim3(B), dim3(BLOCK), 0, stream,
                     pred, tgt, mask, out, B);
}